// StackedLaneGraphConvShared_55722905698617
// MI455X (gfx1250) — compile-verified
//
#include <hip/hip_runtime.h>

#define NN 10000
#define NE 80000
#define D 32
#define DE 16
#define NL 3

typedef __attribute__((ext_vector_type(2))) float v2f;
typedef __attribute__((ext_vector_type(8))) float v8f;

// ---------------------------------------------------------------------------
// Edge kernel: per 16-edge tile (one wave each):
//   h = edge_attr_tile @ emlp_w^T + emlp_b        (WMMA f32 16x16x4, K=16)
//   We = relu(LayerNorm_1024(h))                   (stats via 2-pass recompute)
//   msg[m,f] = sum_d x[src[m],d] * We[m,d,f]       (in-register, col c=32d+f)
//   atomicAdd into agg[tgt]
// ---------------------------------------------------------------------------
__global__ __launch_bounds__(256) void edge_msg_kernel(
    const float* __restrict__ xin,
    const int*   __restrict__ eidx,      // [2, NE]
    const float* __restrict__ eattr,     // [NE, 16]
    const float* __restrict__ emlp_w,    // [1024, 16]
    const float* __restrict__ emlp_b,    // [1024]
    const float* __restrict__ lng,       // [1024]
    const float* __restrict__ lnb,       // [1024]
    float*       __restrict__ agg)       // [NN, 32]
{
    __shared__ __align__(16) float xs[8 * 16 * D];   // 8 waves * (16 edges x 32 feats)

    const int tid  = threadIdx.x;
    const int wave = tid >> 5;
    const int lane = tid & 31;
    const int n    = lane & 15;     // column-in-tile / row M for A
    const int hi   = lane >> 4;     // half-wave selector
    const int tile = blockIdx.x * 8 + wave;
    const int base = tile * 16;

    float* xsw = xs + wave * (16 * D);

    // Stage x[src] rows for this tile: lane (n,hi) loads 16 floats of edge n's src row.
    {
        const int s = eidx[base + n];
        const float* xr = xin + s * D + hi * 16;
        float* dst = xsw + n * D + hi * 16;
#pragma unroll
        for (int j = 0; j < 16; j += 4)
            *(float4*)(dst + j) = *(const float4*)(xr + j);
    }

    // A operands: lane holds A[M=n, K=2*hi+v] for each K-chunk kc (K=4 per WMMA).
    v2f a[4];
    {
        const float* ar = eattr + (base + n) * DE;
#pragma unroll
        for (int kc = 0; kc < 4; ++kc)
            a[kc] = *(const v2f*)(ar + 4 * kc + 2 * hi);
    }

    // ---- pass 1: per-row sum / sum-of-squares over all 1024 columns ----
    float s[8], sq[8];
#pragma unroll
    for (int v = 0; v < 8; ++v) { s[v] = 0.0f; sq[v] = 0.0f; }

    for (int t = 0; t < 64; ++t) {
        const float* wr = emlp_w + (t * 16 + n) * DE;   // B[k,n] = W[16t+n, 4kc+k]
        v8f c = {};
#pragma unroll
        for (int kc = 0; kc < 4; ++kc) {
            v2f b = *(const v2f*)(wr + 4 * kc + 2 * hi);
            c = __builtin_amdgcn_wmma_f32_16x16x4_f32(false, a[kc], false, b,
                                                      (short)0, c, false, false);
        }
        const float bias = emlp_b[t * 16 + n];
#pragma unroll
        for (int v = 0; v < 8; ++v) {
            float h = c[v] + bias;
            s[v]  += h;
            sq[v] += h * h;
        }
    }
    // reduce across the 16 lanes of each half (rows differ between halves)
#pragma unroll
    for (int v = 0; v < 8; ++v) {
#pragma unroll
        for (int m = 8; m >= 1; m >>= 1) {
            s[v]  += __shfl_xor(s[v],  m, 32);
            sq[v] += __shfl_xor(sq[v], m, 32);
        }
    }
    float mean[8], rstd[8];
#pragma unroll
    for (int v = 0; v < 8; ++v) {
        float mu  = s[v] * (1.0f / 1024.0f);
        float var = sq[v] * (1.0f / 1024.0f) - mu * mu;
        mean[v] = mu;
        rstd[v] = rsqrtf(var + 1e-5f);
    }

    // ---- pass 2: recompute tiles, LN + ReLU, accumulate message in registers ----
    float msg0[8], msg1[8];
#pragma unroll
    for (int v = 0; v < 8; ++v) { msg0[v] = 0.0f; msg1[v] = 0.0f; }

    for (int tt = 0; tt < 32; ++tt) {          // tt = d index; tiles 2*tt (f<16), 2*tt+1 (f>=16)
#pragma unroll
        for (int p = 0; p < 2; ++p) {
            const int t = 2 * tt + p;
            const int col = t * 16 + n;
            const float* wr = emlp_w + col * DE;
            v8f c = {};
#pragma unroll
            for (int kc = 0; kc < 4; ++kc) {
                v2f b = *(const v2f*)(wr + 4 * kc + 2 * hi);
                c = __builtin_amdgcn_wmma_f32_16x16x4_f32(false, a[kc], false, b,
                                                          (short)0, c, false, false);
            }
            const float bias = emlp_b[col];
            const float g    = lng[col];
            const float bb   = lnb[col];
#pragma unroll
            for (int v = 0; v < 8; ++v) {
                float h = c[v] + bias;
                float w = (h - mean[v]) * rstd[v] * g + bb;
                w = fmaxf(w, 0.0f);
                float xv = xsw[(v + 8 * hi) * D + tt];   // broadcast within half
                if (p == 0) msg0[v] += xv * w;
                else        msg1[v] += xv * w;
            }
        }
    }

    // scatter-add: lane holds msg[m, n] and msg[m, 16+n] for rows m = v + 8*hi
#pragma unroll
    for (int v = 0; v < 8; ++v) {
        const int m  = v + 8 * hi;
        const int tn = eidx[NE + base + m];
        atomicAdd(agg + tn * D + n,      msg0[v]);
        atomicAdd(agg + tn * D + 16 + n, msg1[v]);
    }
}

// ---------------------------------------------------------------------------
// agg = x @ root_w^T + conv_b ; xres = x @ res_w^T + res_b   (one thread per (node,f))
// ---------------------------------------------------------------------------
__global__ __launch_bounds__(256) void node_pre_kernel(
    const float* __restrict__ xin,
    const float* __restrict__ root_w,   // [32,32] this layer
    const float* __restrict__ conv_b,   // [32]
    const float* __restrict__ res_w,
    const float* __restrict__ res_b,
    float* __restrict__ agg,
    float* __restrict__ xres)
{
    const int idx  = blockIdx.x * 256 + threadIdx.x;
    const int node = idx >> 5;
    const int f    = idx & 31;
    if (node >= NN) return;
    float aR = conv_b[f];
    float aS = res_b[f];
    const float* xr = xin + node * D;
    const float* wr = root_w + f * D;
    const float* sr = res_w + f * D;
#pragma unroll 8
    for (int d = 0; d < D; ++d) {
        float xv = xr[d];
        aR = fmaf(xv, wr[d], aR);
        aS = fmaf(xv, sr[d], aS);
    }
    agg[node * D + f]  = aR;
    xres[node * D + f] = aS;
}

// ---------------------------------------------------------------------------
// x_out = LayerNorm_32(agg + xres) * g + b    (one wave per node)
// ---------------------------------------------------------------------------
__global__ __launch_bounds__(256) void node_post_kernel(
    const float* __restrict__ agg,
    const float* __restrict__ xres,
    const float* __restrict__ g,
    const float* __restrict__ b,
    float* __restrict__ xout)
{
    const int idx  = blockIdx.x * 256 + threadIdx.x;
    const int node = idx >> 5;
    const int f    = idx & 31;
    if (node >= NN) return;
    float h = agg[node * D + f] + xres[node * D + f];
    float s = h, sq = h * h;
#pragma unroll
    for (int m = 16; m >= 1; m >>= 1) {
        s  += __shfl_xor(s,  m, 32);
        sq += __shfl_xor(sq, m, 32);
    }
    float mu  = s * (1.0f / 32.0f);
    float var = sq * (1.0f / 32.0f) - mu * mu;
    float o = (h - mu) * rsqrtf(var + 1e-5f) * g[f] + b[f];
    xout[node * D + f] = o;
}

// ---------------------------------------------------------------------------
// fused = relu(LayerNorm_32(concat(o0,o1,o2) @ fuse_w^T + fuse_b))
// ---------------------------------------------------------------------------
__global__ __launch_bounds__(256) void fuse_kernel(
    const float* __restrict__ o0,
    const float* __restrict__ o1,
    const float* __restrict__ o2,
    const float* __restrict__ fw,    // [32, 96]
    const float* __restrict__ fb,
    const float* __restrict__ g,
    const float* __restrict__ b,
    float* __restrict__ out)
{
    const int idx  = blockIdx.x * 256 + threadIdx.x;
    const int node = idx >> 5;
    const int f    = idx & 31;
    if (node >= NN) return;
    float acc = fb[f];
    const float* wr = fw + f * (NL * D);
    const float* rows[3] = { o0 + node * D, o1 + node * D, o2 + node * D };
#pragma unroll
    for (int li = 0; li < 3; ++li)
#pragma unroll 8
        for (int d = 0; d < D; ++d)
            acc = fmaf(rows[li][d], wr[li * D + d], acc);
    float s = acc, sq = acc * acc;
#pragma unroll
    for (int m = 16; m >= 1; m >>= 1) {
        s  += __shfl_xor(s,  m, 32);
        sq += __shfl_xor(sq, m, 32);
    }
    float mu  = s * (1.0f / 32.0f);
    float var = sq * (1.0f / 32.0f) - mu * mu;
    float o = (acc - mu) * rsqrtf(var + 1e-5f) * g[f] + b[f];
    out[node * D + f] = fmaxf(o, 0.0f);
}

extern "C" void kernel_launch(void* const* d_in, const int* in_sizes, int n_in,
                              void* d_out, int out_size, void* d_ws, size_t ws_size,
                              hipStream_t stream) {
    const float* x       = (const float*)d_in[0];
    const int*   eidx    = (const int*)  d_in[1];
    const float* eattr   = (const float*)d_in[2];
    const float* emlp_w  = (const float*)d_in[3];
    const float* emlp_b  = (const float*)d_in[4];
    const float* emlp_g  = (const float*)d_in[5];
    const float* emlp_bb = (const float*)d_in[6];
    const float* root_w  = (const float*)d_in[7];
    const float* conv_b  = (const float*)d_in[8];
    const float* res_w   = (const float*)d_in[9];
    const float* res_b   = (const float*)d_in[10];
    const float* ln_g    = (const float*)d_in[11];
    const float* ln_b    = (const float*)d_in[12];
    const float* fuse_w  = (const float*)d_in[13];
    const float* fuse_b  = (const float*)d_in[14];
    const float* fuse_g  = (const float*)d_in[15];
    const float* fuse_bb = (const float*)d_in[16];

    float* ws   = (float*)d_ws;
    float* agg  = ws;                       // [NN,32]
    float* xres = ws + (size_t)NN * D;      // [NN,32]
    float* outs[NL] = { ws + 2 * (size_t)NN * D,
                        ws + 3 * (size_t)NN * D,
                        ws + 4 * (size_t)NN * D };

    const int nodeBlocks = (NN * D + 255) / 256;   // 1250
    const int edgeBlocks = NE / 16 / 8;            // 625 (8 waves/block, 16 edges/wave)

    const float* xcur = x;
    for (int l = 0; l < NL; ++l) {
        node_pre_kernel<<<nodeBlocks, 256, 0, stream>>>(
            xcur, root_w + l * D * D, conv_b + l * D,
            res_w + l * D * D, res_b + l * D, agg, xres);
        edge_msg_kernel<<<edgeBlocks, 256, 0, stream>>>(
            xcur, eidx, eattr, emlp_w, emlp_b, emlp_g, emlp_bb, agg);
        node_post_kernel<<<nodeBlocks, 256, 0, stream>>>(
            agg, xres, ln_g + l * D, ln_b + l * D, outs[l]);
        xcur = outs[l];
    }
    fuse_kernel<<<nodeBlocks, 256, 0, stream>>>(
        outs[0], outs[1], outs[2], fuse_w, fuse_b, fuse_g, fuse_bb, (float*)d_out);
}